// FreezedScaledFakeQuantize_46815143526621
// MI455X (gfx1250) — compile-verified
//
#include <hip/hip_runtime.h>

// Fake-quantize (per-row, group-size-128) on 8192x8192 fp32 — pure streaming op.
// Roofline: 516 MiB traffic @ 23.3 TB/s -> ~22 us; goal is to keep VALU/SALU
// comfortably below that. Wave32 mapping: 32 lanes x float4 = 128 cols = one
// quant group -> scale/zero_point are wave-uniform scalar loads.
//
// VALU diet vs round 1:
//  - one IEEE reciprocal per row + Markstein fma refinement gives the
//    correctly-rounded quotient RN(x/s) in 3 ops/element (bit-exact with
//    reference division; all operands normal-range here),
//  - v_med3_f32 single-op clamp,
//  - 4 rows/thread amortizes the per-wave SALU prologue and quadruples
//    memory-level parallelism (4 independent NT b128 loads in flight).

typedef float v4f __attribute__((ext_vector_type(4)));

#define ROWS    8192
#define COLS    8192
#define NGROUPS 64      // COLS / 128
#define RPT     4       // rows per thread

__global__ __launch_bounds__(256) void fakequant_g128_kernel(
    const float* __restrict__ X,
    const float* __restrict__ scale,
    const float* __restrict__ zero_point,
    const int*   __restrict__ qmin_p,
    const int*   __restrict__ qmax_p,
    float*       __restrict__ out)
{
    const float qmin = (float)qmin_p[0];
    const float qmax = (float)qmax_p[0];

    const int row0 = blockIdx.y * RPT;
    const int col  = (blockIdx.x * 256 + threadIdx.x) * 4;   // 4 floats per lane

    // Wave-uniform group index (one wave32 covers exactly one 128-col group);
    // readfirstlane makes scale/zp addresses scalar -> s_load_b32.
    const int g = __builtin_amdgcn_readfirstlane(col >> 7);

    const size_t base = (size_t)row0 * COLS + (size_t)col;

    // Kick off all 4 NT streaming loads up front (4-deep MLP per lane).
    v4f x[RPT];
#pragma unroll
    for (int r = 0; r < RPT; ++r)
        x[r] = __builtin_nontemporal_load((const v4f*)(X + base + (size_t)r * COLS));

    float s[RPT], z[RPT], c[RPT];
#pragma unroll
    for (int r = 0; r < RPT; ++r) {
        s[r] = scale[(size_t)(row0 + r) * NGROUPS + g];       // scalar load
        z[r] = zero_point[(size_t)(row0 + r) * NGROUPS + g];  // scalar load
        c[r] = 1.0f / s[r];   // IEEE correctly-rounded reciprocal (once per row)
    }

#pragma unroll
    for (int r = 0; r < RPT; ++r) {
        v4f q;
#pragma unroll
        for (int i = 0; i < 4; ++i) {
            // Markstein refinement: with c = RN(1/s), this yields RN(x/s)
            // (correctly rounded, matches the reference's exact division).
            float q0 = x[r][i] * c[r];
            float rr = fmaf(-s[r], q0, x[r][i]);   // exact remainder via fma
            float d  = fmaf(rr, c[r], q0);         // == RN(x/s)
            // RNE round (jnp.round), single-op clamp, dequantize in
            // reference order: (t - zp) * s.
            float t  = __builtin_rintf(d + z[r]);
            t = __builtin_amdgcn_fmed3f(t, qmin, qmax);
            q[i] = (t - z[r]) * s[r];
        }
        // Streamed output, never re-read -> NT store keeps L2 clean.
        __builtin_nontemporal_store(q, (v4f*)(out + base + (size_t)r * COLS));
    }
}

extern "C" void kernel_launch(void* const* d_in, const int* in_sizes, int n_in,
                              void* d_out, int out_size, void* d_ws, size_t ws_size,
                              hipStream_t stream) {
    const float* X     = (const float*)d_in[0];
    const float* scale = (const float*)d_in[1];
    const float* zp    = (const float*)d_in[2];
    const int*   qmin  = (const int*)d_in[3];
    const int*   qmax  = (const int*)d_in[4];
    float*       out   = (float*)d_out;

    (void)in_sizes; (void)n_in; (void)out_size; (void)d_ws; (void)ws_size;

    dim3 block(256);                                  // 8 wave32s
    dim3 grid(COLS / (256 * 4), ROWS / RPT);          // 8 x 2048 blocks
    fakequant_g128_kernel<<<grid, block, 0, stream>>>(X, scale, zp, qmin, qmax, out);
}